// THNN_layer_90185723281669
// MI455X (gfx1250) — compile-verified
//
#include <hip/hip_runtime.h>
#include <hip/hip_bf16.h>
#include <cstdint>

// ---------------------------------------------------------------------------
// THNN hypergraph layer, MI455X (gfx1250, wave32, WMMA + async Global->LDS).
// fp32 end-to-end; matrix ops via V_WMMA_F32_16X16X4_F32.
// ---------------------------------------------------------------------------

#define FEAT 128
#define HID  256
#define OUTD 128
#define RANK 50
#define KMEM 4
#define KPAD 132   // FEAT+1 padded to multiple of 4
#define RPAD 64    // RANK padded to multiple of 16

typedef __attribute__((ext_vector_type(2))) float v2f;
typedef __attribute__((ext_vector_type(8))) float v8f;
typedef int v4i_vs __attribute__((vector_size(16)));   // matches builtin's b128 pointee

// ---- gfx1250 async Global->LDS copy path (ASYNCcnt), with safe fallback ----
#if defined(__has_builtin)
#if __has_builtin(__builtin_amdgcn_global_load_async_to_lds_b32)
#define HAVE_ASYNC_LDS 1
#endif
#endif

#if defined(HAVE_ASYNC_LDS)
// Generic "shared" pointer -> LDS offset: flat shared addr keeps the LDS byte
// offset in ADDR[31:0] (aperture in the high half), so truncate + reinterpret.
#define GLB_I32(p)  ((__attribute__((address_space(1))) int*)(uintptr_t)(p))
#define LDS_I32(p)  ((__attribute__((address_space(3))) int*)(uint32_t)(uintptr_t)(p))
#define GLB_I128(p) ((__attribute__((address_space(1))) v4i_vs*)(uintptr_t)(p))
#define LDS_I128(p) ((__attribute__((address_space(3))) v4i_vs*)(uint32_t)(uintptr_t)(p))
#define ASYNC_CP_B32(dst, src)  __builtin_amdgcn_global_load_async_to_lds_b32(GLB_I32(src),  LDS_I32(dst),  0, 0)
#define ASYNC_CP_B128(dst, src) __builtin_amdgcn_global_load_async_to_lds_b128(GLB_I128(src), LDS_I128(dst), 0, 0)
#if __has_builtin(__builtin_amdgcn_s_wait_asynccnt)
#define WAIT_ASYNC() __builtin_amdgcn_s_wait_asynccnt(0)
#else
#define WAIT_ASYNC() asm volatile("s_wait_asynccnt 0x0" ::: "memory")
#endif
#else
#define ASYNC_CP_B32(dst, src)  (*(float*)(dst) = *(const float*)(src))
#define ASYNC_CP_B128(dst, src) (*(float4*)(dst) = *(const float4*)(src))
#define WAIT_ASYNC() ((void)0)
#endif

// ----------------------------- prep kernels --------------------------------

__global__ void prep_emb1(const float* __restrict__ emb, float* __restrict__ emb1p, int n_nodes) {
    int i = blockIdx.x * blockDim.x + threadIdx.x;
    int total = n_nodes * KPAD;
    if (i >= total) return;
    int n = i / KPAD, c = i % KPAD;
    float v;
    if (c < FEAT)       v = emb[n * FEAT + c];
    else if (c == FEAT) v = 1.0f;
    else                v = 0.0f;
    emb1p[i] = v;
}

__global__ void pad_matrix(const float* __restrict__ W, float* __restrict__ Wp,
                           int rows, int cols, int prows, int pcols) {
    int i = blockIdx.x * blockDim.x + threadIdx.x;
    int total = prows * pcols;
    if (i >= total) return;
    int rr = i / pcols, cc = i % pcols;
    Wp[i] = (rr < rows && cc < cols) ? W[rr * cols + cc] : 0.0f;
}

__global__ void pad_vec(const float* __restrict__ b, float* __restrict__ bp, int n, int np) {
    int i = blockIdx.x * blockDim.x + threadIdx.x;
    if (i < np) bp[i] = (i < n) ? b[i] : 0.0f;
}

// ------------------------- degree / scale kernels --------------------------

__global__ void degree_kernel(const long long* __restrict__ edges, float* __restrict__ deg, int n_entries) {
    int i = blockIdx.x * blockDim.x + threadIdx.x;
    if (i < n_entries) atomicAdd(&deg[(int)edges[i]], 1.0f);
}

__global__ void scale_kernel(const float* __restrict__ deg, float* __restrict__ scale, int n_nodes) {
    int i = blockIdx.x * blockDim.x + threadIdx.x;
    if (i < n_nodes) scale[i] = powf(deg[i], 0.25f);   // deg^(1/K), K=4
}

// ---------------------- generic fp32 WMMA GEMM kernel ----------------------
// One block = one 16-row strip of C. A strip (16 x K, contiguous rows) is
// async-copied to LDS once (odd stride K+1 -> conflict-free frag reads), then
// the 4 waves split the column tiles. B fragments come from global (small,
// L2/L1 resident, reused across all 6250 blocks).
// Requires: M%16==0, Ncols%16==0, K%4==0, lda==K.
// WMMA f32 16x16x4 fragment layout (per lane; half=lane/16, r=lane%16):
//   A: a[v] = A[m0+r][k0 + 2*half + v]
//   B: b[v] = B[k0 + 2*half + v][n0+r]
//   D: d[v] = C[m0 + v + 8*half][n0+r]
__global__ void __launch_bounds__(128)
gemm_f32_wmma(const float* __restrict__ A, const float* __restrict__ B,
              const float* __restrict__ bias, float* __restrict__ C,
              int Ncols, int K, int ldb, int ldc, int do_relu) {
    extern __shared__ float sA[];          // [16][K+1]
    const int tid    = threadIdx.x;
    const int rt     = blockIdx.x;
    const int stride = K + 1;

    // stage A strip into LDS via async Global->LDS copies
    {
        const float* Ab = A + (size_t)rt * 16 * K;
        const int total = 16 * K;
        for (int t = tid; t < total; t += 128) {
            int row = t / K, k = t - row * K;
            ASYNC_CP_B32(&sA[row * stride + k], &Ab[(size_t)row * K + k]);
        }
        WAIT_ASYNC();
    }
    __syncthreads();

    const int lane = tid & 31;
    const int wave = tid >> 5;
    const int half = lane >> 4;
    const int r    = lane & 15;
    const int ntc  = Ncols >> 4;
    const float* Arow = &sA[r * stride];

    for (int ct = wave; ct < ntc; ct += 4) {
        const int n = ct * 16 + r;
        const float* Bcol = B + n;
        v8f acc = {};
        for (int k0 = 0; k0 < K; k0 += 4) {
            v2f a, b;
            a[0] = Arow[k0 + 2 * half + 0];
            a[1] = Arow[k0 + 2 * half + 1];
            b[0] = Bcol[(size_t)(k0 + 2 * half + 0) * ldb];
            b[1] = Bcol[(size_t)(k0 + 2 * half + 1) * ldb];
            acc = __builtin_amdgcn_wmma_f32_16x16x4_f32(false, a, false, b, (short)0, acc, false, false);
        }
        const float bv = bias[n];
#pragma unroll
        for (int v = 0; v < 8; ++v) {
            float o = acc[v] + bv;
            if (do_relu) o = fmaxf(o, 0.0f);
            C[(size_t)(rt * 16 + v + 8 * half) * ldc + n] = o;
        }
    }
}

// ----------------------------- edge kernel ---------------------------------
// 16 edges (= 64 rows of the [E*K, RPAD] implicit matrix) per 128-thread block.
// Phase 0: async-stage the full padded W_q (RPADxOUTD) into LDS (ASYNCcnt),
//          while gathering edge member ids + scales.
// Phase 1: gather g = scale*emb_p (float4, L2-resident table) into LDS.
// Phase 2: exclusive-product + tanh A-tile into LDS; relu'd edge sums in LDS.
// Phase 3: WMMA q-GEMM, both operands from LDS; epilogue adds b_q + relu(sum2)
//          and scatter-adds into sums with global f32 atomics.
#define EPB 16

__global__ void __launch_bounds__(128)
edge_kernel(const long long* __restrict__ edges,
            const float* __restrict__ embp,   // [N, RPAD], cols >= RANK are 0
            const float* __restrict__ embp2,  // [N, OUTD]
            const float* __restrict__ scale,  // [N]
            const float* __restrict__ Wq,     // [RPAD, OUTD], rows >= RANK are 0
            const float* __restrict__ bq,     // [OUTD]
            float* __restrict__ sums) {       // [N, OUTD]
    __shared__ int   sIdx[64];
    __shared__ float sS[64];
    __shared__ __align__(16) float sG[64][68];    // stride 68: conflict-free frag reads
    __shared__ __align__(16) float sA[64][68];
    __shared__ __align__(16) float sES[EPB][OUTD];
    __shared__ __align__(16) float sWq[RPAD][OUTD + 4];  // stride 132

    const int tid = threadIdx.x;
    const int e0  = blockIdx.x * EPB;

    // async-stage W_q into LDS (B128 = 4 floats per copy); completion is
    // enforced per-wave by WAIT_ASYNC before the last barrier below.
    for (int t = tid; t < RPAD * (OUTD / 4); t += 128) {
        int row = t >> 5, c4 = t & 31;                  // OUTD/4 == 32
        ASYNC_CP_B128(&sWq[row][c4 * 4], &Wq[(size_t)row * OUTD + c4 * 4]);
    }

    if (tid < 64) {
        int node = (int)edges[(size_t)e0 * KMEM + tid];
        sIdx[tid] = node;
        sS[tid]   = scale[node];
    }
    __syncthreads();

    // g[row][c] = scale[node] * emb_p[node][c]   (vectorized gather)
    for (int t = tid; t < 64 * (RPAD / 4); t += 128) {
        int row = t >> 4, c4 = t & 15;                  // RPAD/4 == 16
        float4 v = *(const float4*)&embp[(size_t)sIdx[row] * RPAD + c4 * 4];
        float s = sS[row];
        v.x *= s; v.y *= s; v.z *= s; v.w *= s;
        *(float4*)&sG[row][c4 * 4] = v;
    }
    __syncthreads();

    // A[row][c] = tanh(scale_i/3! * prod_{j != i} g_j[c])
    const float inv6 = 1.0f / 6.0f;
    for (int t = tid; t < 64 * RPAD; t += 128) {
        int row = t >> 6, c = t & 63;
        int base = row & ~3;
        int mi   = row & 3;
        float p = 1.0f;
#pragma unroll
        for (int j = 0; j < 4; ++j)
            if (j != mi) p *= sG[base + j][c];
        sA[row][c] = tanhf(sS[row] * inv6 * p);
    }

    // relu(edge_sum2) (vectorized gather; independent of sA)
    for (int t = tid; t < EPB * (OUTD / 4); t += 128) {
        int el = t >> 5, c4 = t & 31;                   // OUTD/4 == 32
        int b4 = el * 4;
        const float4 a0 = *(const float4*)&embp2[(size_t)sIdx[b4 + 0] * OUTD + c4 * 4];
        const float4 a1 = *(const float4*)&embp2[(size_t)sIdx[b4 + 1] * OUTD + c4 * 4];
        const float4 a2 = *(const float4*)&embp2[(size_t)sIdx[b4 + 2] * OUTD + c4 * 4];
        const float4 a3 = *(const float4*)&embp2[(size_t)sIdx[b4 + 3] * OUTD + c4 * 4];
        float4 s;
        s.x = fmaxf(a0.x + a1.x + a2.x + a3.x, 0.0f);
        s.y = fmaxf(a0.y + a1.y + a2.y + a3.y, 0.0f);
        s.z = fmaxf(a0.z + a1.z + a2.z + a3.z, 0.0f);
        s.w = fmaxf(a0.w + a1.w + a2.w + a3.w, 0.0f);
        *(float4*)&sES[el][c4 * 4] = s;
    }
    WAIT_ASYNC();          // our wave's W_q async copies are done
    __syncthreads();       // -> all waves' copies + LDS tiles visible

    // WMMA: wave w owns rows [16w, 16w+16); loop over the 8 column tiles.
    const int lane = tid & 31;
    const int wave = tid >> 5;
    const int half = lane >> 4;
    const int r    = lane & 15;
    const int rowbase = wave * 16;

    for (int ct = 0; ct < OUTD / 16; ++ct) {
        const int n = ct * 16 + r;
        v8f acc = {};
        for (int k0 = 0; k0 < RPAD; k0 += 4) {
            v2f a, b;
            a[0] = sA[rowbase + r][k0 + 2 * half + 0];
            a[1] = sA[rowbase + r][k0 + 2 * half + 1];
            b[0] = sWq[k0 + 2 * half + 0][n];
            b[1] = sWq[k0 + 2 * half + 1][n];
            acc = __builtin_amdgcn_wmma_f32_16x16x4_f32(false, a, false, b, (short)0, acc, false, false);
        }
        const float bb = bq[n];
#pragma unroll
        for (int v = 0; v < 8; ++v) {
            int m = rowbase + v + 8 * half;
            float val = acc[v] + bb + sES[m >> 2][n];
            atomicAdd(&sums[(size_t)sIdx[m] * OUTD + n], val);
        }
    }
}

// ----------------------------- finalize ------------------------------------

__global__ void finalize_kernel(const float* __restrict__ sums, const float* __restrict__ deg,
                                const float* __restrict__ residual, float* __restrict__ out,
                                int total) {
    int i = blockIdx.x * blockDim.x + threadIdx.x;
    if (i >= total) return;
    int n = i >> 7;  // OUTD == 128
    float d = fmaxf(deg[n], 1.0f);
    out[i] = fmaxf(sums[i] / d, 0.0f) + residual[i];
}

// ----------------------------- launch --------------------------------------

extern "C" void kernel_launch(void* const* d_in, const int* in_sizes, int n_in,
                              void* d_out, int out_size, void* d_ws, size_t ws_size,
                              hipStream_t stream) {
    const float*     embedding = (const float*)d_in[0];
    const long long* edges     = (const long long*)d_in[1];
    const float*     W_p   = (const float*)d_in[2];
    const float*     b_p   = (const float*)d_in[3];
    const float*     W_q   = (const float*)d_in[4];
    const float*     b_q   = (const float*)d_in[5];
    const float*     W_p2a = (const float*)d_in[6];
    const float*     b_p2a = (const float*)d_in[7];
    const float*     W_p2b = (const float*)d_in[8];
    const float*     b_p2b = (const float*)d_in[9];
    const float*     W_al  = (const float*)d_in[10];
    const float*     b_al  = (const float*)d_in[11];
    float* out = (float*)d_out;

    const int N = in_sizes[0] / FEAT;          // 100000
    const int E = in_sizes[1] / KMEM;          // 400000
    const int EK = E * KMEM;

    // workspace bump allocator (floats); every block stays 16B aligned
    float* ws = (float*)d_ws;
    size_t off = 0;
    float* emb1p    = ws + off; off += (size_t)N * KPAD;
    float* residual = ws + off; off += (size_t)N * OUTD;
    float* embp     = ws + off; off += (size_t)N * RPAD;
    float* hbuf     = ws + off; off += (size_t)N * HID;
    float* embp2    = ws + off; off += (size_t)N * OUTD;
    float* deg      = ws + off; off += (size_t)N;
    float* scal     = ws + off; off += (size_t)N;
    float* sums     = ws + off; off += (size_t)N * OUTD;
    float* Wal_p    = ws + off; off += (size_t)KPAD * OUTD;
    float* Wp_p     = ws + off; off += (size_t)KPAD * RPAD;
    float* Wp2a_p   = ws + off; off += (size_t)KPAD * HID;
    float* Wq_p     = ws + off; off += (size_t)RPAD * OUTD;
    float* bp_p     = ws + off; off += (size_t)RPAD;
    (void)ws_size; (void)n_in; (void)out_size;

    // zero the accumulators (graph-capture safe)
    (void)hipMemsetAsync(deg,  0, (size_t)N * sizeof(float), stream);
    (void)hipMemsetAsync(sums, 0, (size_t)N * OUTD * sizeof(float), stream);

    // prep: padded A operand and padded weights
    {
        int t = N * KPAD;
        prep_emb1<<<(t + 255) / 256, 256, 0, stream>>>(embedding, emb1p, N);
    }
    pad_matrix<<<((KPAD * OUTD) + 255) / 256, 256, 0, stream>>>(W_al,  Wal_p,  FEAT + 1, OUTD, KPAD, OUTD);
    pad_matrix<<<((KPAD * RPAD) + 255) / 256, 256, 0, stream>>>(W_p,   Wp_p,   FEAT + 1, RANK, KPAD, RPAD);
    pad_matrix<<<((KPAD * HID ) + 255) / 256, 256, 0, stream>>>(W_p2a, Wp2a_p, FEAT + 1, HID,  KPAD, HID);
    pad_matrix<<<((RPAD * OUTD) + 255) / 256, 256, 0, stream>>>(W_q,   Wq_p,   RANK,     OUTD, RPAD, OUTD);
    pad_vec<<<1, 64, 0, stream>>>(b_p, bp_p, RANK, RPAD);

    // degree + scale
    degree_kernel<<<(EK + 255) / 256, 256, 0, stream>>>(edges, deg, EK);
    scale_kernel<<<(N + 255) / 256, 256, 0, stream>>>(deg, scal, N);

    // node-side GEMMs: one 16-row strip per block, A staged via async LDS copy
    const int mt = N / 16;
    const size_t sh132 = (size_t)16 * (KPAD + 1) * sizeof(float);
    const size_t sh256 = (size_t)16 * (HID  + 1) * sizeof(float);
    // residual = relu(emb1 @ W_al + b_al)
    gemm_f32_wmma<<<mt, 128, sh132, stream>>>(emb1p, Wal_p,  b_al,  residual, OUTD, KPAD, OUTD, OUTD, 1);
    // emb_p = emb1 @ W_p + b_p   (padded to RPAD cols)
    gemm_f32_wmma<<<mt, 128, sh132, stream>>>(emb1p, Wp_p,   bp_p,  embp,     RPAD, KPAD, RPAD, RPAD, 0);
    // h = relu(emb1 @ W_p2a + b_p2a)
    gemm_f32_wmma<<<mt, 128, sh132, stream>>>(emb1p, Wp2a_p, b_p2a, hbuf,     HID,  KPAD, HID,  HID,  1);
    // emb_p2 = h @ W_p2b + b_p2b   (K=256 already WMMA-friendly, raw weights)
    gemm_f32_wmma<<<mt, 128, sh256, stream>>>(hbuf,  W_p2b,  b_p2b, embp2,    OUTD, HID,  OUTD, OUTD, 0);

    // fused edge pipeline: gather -> exclusive product -> tanh -> WMMA q_net
    // -> + b_q + relu(edge_sum2) -> scatter-add into sums
    edge_kernel<<<E / EPB, 128, 0, stream>>>(edges, embp, embp2, scal, Wq_p, b_q, sums);

    // out = relu(sums / max(deg,1)) + residual
    finalize_kernel<<<((N * OUTD) + 255) / 256, 256, 0, stream>>>(sums, deg, residual, out, N * OUTD);
}